// GatGSO_26852135535022
// MI455X (gfx1250) — compile-verified
//
#include <hip/hip_runtime.h>

// Shapes from the reference
#define B_  16
#define G_  128
#define N_  512
#define P_  4
#define KH_ 3      // hops
#define F_  128
#define BP_ (B_*P_)

#define NEG_SLOPE 0.2f
#define ZTOL      1e-9f
#define NEG_INF12 (-1.0e12f)

typedef float v2f __attribute__((ext_vector_type(2)));
typedef float v8f __attribute__((ext_vector_type(8)));

// Exact f32 WMMA: D(16x16) = A(16x4) * B(4x16) + C
__device__ __forceinline__ v8f wmma4(v2f a, v2f b, v8f c) {
  return __builtin_amdgcn_wmma_f32_16x16x4_f32(
      /*neg_a=*/false, a, /*neg_b=*/false, b,
      /*c_mod=*/(short)0, c, /*reuse_a=*/false, /*reuse_b=*/false);
}

// ---------------------------------------------------------------------------
// Kernel 1: Wx[bp] = W[p] (GxG) @ x[b] (GxN)
// wave computes a 32x32 C block (2x2 WMMA tiles). grid(4, 4, 64) block 128
// ---------------------------------------------------------------------------
__global__ __launch_bounds__(128) void k_wx(const float* __restrict__ x,
                                            const float* __restrict__ W,
                                            float* __restrict__ wx) {
  const int lane = threadIdx.x & 31, wave = threadIdx.x >> 5;
  const int bp = blockIdx.z, b = bp >> 2, p = bp & 3;
  const int tm = blockIdx.x, tn = blockIdx.y * 4 + wave;
  const int lm = lane & 15, kh = (lane >> 4) * 2, khalf = lane >> 4;
  const int m0 = tm * 32 + lm, m1 = m0 + 16;
  const int n0 = tn * 32 + lm, n1 = n0 + 16;

  const float* __restrict__ A = W + (size_t)p * G_ * G_;   // row-major (G,G)
  const float* __restrict__ Bm = x + (size_t)b * G_ * N_;  // row-major (G,N)

  v8f c00 = {}, c01 = {}, c10 = {}, c11 = {};
  for (int kk = 0; kk < G_; kk += 4) {
    const int k0 = kk + kh;
    v2f a0; a0.x = A[m0 * G_ + k0]; a0.y = A[m0 * G_ + k0 + 1];
    v2f a1; a1.x = A[m1 * G_ + k0]; a1.y = A[m1 * G_ + k0 + 1];
    v2f b0; b0.x = Bm[k0 * N_ + n0]; b0.y = Bm[(k0 + 1) * N_ + n0];
    v2f b1; b1.x = Bm[k0 * N_ + n1]; b1.y = Bm[(k0 + 1) * N_ + n1];
    c00 = wmma4(a0, b0, c00); c01 = wmma4(a0, b1, c01);
    c10 = wmma4(a1, b0, c10); c11 = wmma4(a1, b1, c11);
  }
  float* __restrict__ C = wx + (size_t)bp * G_ * N_;
#pragma unroll
  for (int v = 0; v < 8; ++v) {
    const int r0 = tm * 32 + v + 8 * khalf, r1 = r0 + 16;
    C[(size_t)r0 * N_ + n0] = c00[v];
    C[(size_t)r0 * N_ + n1] = c01[v];
    C[(size_t)r1 * N_ + n0] = c10[v];
    C[(size_t)r1 * N_ + n1] = c11[v];
  }
}

// ---------------------------------------------------------------------------
// Kernel 2: scores[bp][m][n] = leakyrelu( x[b]^T @ Wx[bp] ), masked by |S|>tol
// 32x32 block per wave. grid(16, 4, 64) block 128
// ---------------------------------------------------------------------------
__global__ __launch_bounds__(128) void k_scores(const float* __restrict__ x,
                                                const float* __restrict__ wx,
                                                const float* __restrict__ S,
                                                float* __restrict__ scores) {
  const int lane = threadIdx.x & 31, wave = threadIdx.x >> 5;
  const int bp = blockIdx.z, b = bp >> 2;
  const int tm = blockIdx.x, tn = blockIdx.y * 4 + wave;
  const int lm = lane & 15, kh = (lane >> 4) * 2, khalf = lane >> 4;
  const int m0 = tm * 32 + lm, m1 = m0 + 16;
  const int n0 = tn * 32 + lm, n1 = n0 + 16;

  const float* __restrict__ xb = x + (size_t)b * G_ * N_;   // A[m][g] = xb[g*N + m]
  const float* __restrict__ Bm = wx + (size_t)bp * G_ * N_; // (G,N)

  v8f c00 = {}, c01 = {}, c10 = {}, c11 = {};
  for (int kk = 0; kk < G_; kk += 4) {
    const int k0 = kk + kh;
    v2f a0; a0.x = xb[k0 * N_ + m0]; a0.y = xb[(k0 + 1) * N_ + m0];
    v2f a1; a1.x = xb[k0 * N_ + m1]; a1.y = xb[(k0 + 1) * N_ + m1];
    v2f b0; b0.x = Bm[k0 * N_ + n0]; b0.y = Bm[(k0 + 1) * N_ + n0];
    v2f b1; b1.x = Bm[k0 * N_ + n1]; b1.y = Bm[(k0 + 1) * N_ + n1];
    c00 = wmma4(a0, b0, c00); c01 = wmma4(a0, b1, c01);
    c10 = wmma4(a1, b0, c10); c11 = wmma4(a1, b1, c11);
  }
  float* __restrict__ out = scores + (size_t)bp * N_ * N_;
  const float* __restrict__ Sb = S + (size_t)b * N_ * N_;   // E == 1
#pragma unroll
  for (int v = 0; v < 8; ++v) {
    const int r0 = tm * 32 + v + 8 * khalf, r1 = r0 + 16;
    const float e[4] = { c00[v], c01[v], c10[v], c11[v] };
    const int   mr[4] = { r0, r0, r1, r1 };
    const int   nc[4] = { n0, n1, n0, n1 };
#pragma unroll
    for (int q = 0; q < 4; ++q) {
      float ev = e[q];
      ev = ev > 0.f ? ev : NEG_SLOPE * ev;                       // leaky relu
      const bool msk = fabsf(Sb[(size_t)mr[q] * N_ + nc[q]]) > ZTOL;
      out[(size_t)mr[q] * N_ + nc[q]] = msk ? ev : NEG_INF12;
    }
  }
}

// ---------------------------------------------------------------------------
// Kernel 3: in-place row softmax over n (512), then zero masked entries.
// One 256-thread block per row (2 elements/thread). grid(BP_*N_)
// ---------------------------------------------------------------------------
__global__ __launch_bounds__(256) void k_softmax(float* __restrict__ aij,
                                                 const float* __restrict__ S) {
  const int row = blockIdx.x;          // row = bp*N + m
  const int bp = row >> 9;             // /N_
  const int m  = row & (N_ - 1);
  const int b  = bp >> 2;
  float* __restrict__ rp = aij + (size_t)row * N_;
  const float* __restrict__ sr = S + ((size_t)b * N_ + m) * N_;

  const int t = threadIdx.x;
  const float s0 = rp[t], s1 = rp[t + 256];

  __shared__ float red[8];

  // --- max reduction ---
  float mx = fmaxf(s0, s1);
#pragma unroll
  for (int off = 16; off; off >>= 1) mx = fmaxf(mx, __shfl_xor(mx, off, 32));
  if ((t & 31) == 0) red[t >> 5] = mx;
  __syncthreads();
  mx = red[0];
#pragma unroll
  for (int i = 1; i < 8; ++i) mx = fmaxf(mx, red[i]);
  __syncthreads();

  // --- sum reduction ---
  const float e0 = __expf(s0 - mx), e1 = __expf(s1 - mx);
  float sm = e0 + e1;
#pragma unroll
  for (int off = 16; off; off >>= 1) sm += __shfl_xor(sm, off, 32);
  if ((t & 31) == 0) red[t >> 5] = sm;
  __syncthreads();
  sm = red[0];
#pragma unroll
  for (int i = 1; i < 8; ++i) sm += red[i];

  const float inv = 1.0f / sm;
  rp[t]       = (fabsf(sr[t])       > ZTOL) ? e0 * inv : 0.f;
  rp[t + 256] = (fabsf(sr[t + 256]) > ZTOL) ? e1 * inv : 0.f;
}

// ---------------------------------------------------------------------------
// Kernel 4: hop GEMM  zout[bp] (GxN) = zin (GxN) @ aij[bp] (NxN)
// 32x32 block per wave. grid(4, 4, 64) block 128
// ---------------------------------------------------------------------------
__global__ __launch_bounds__(128) void k_hop(const float* __restrict__ zin,
                                             long zin_stride, int use_b,
                                             const float* __restrict__ aij,
                                             float* __restrict__ zout,
                                             long zout_stride) {
  const int lane = threadIdx.x & 31, wave = threadIdx.x >> 5;
  const int bp = blockIdx.z, b = bp >> 2;
  const int tm = blockIdx.x, tn = blockIdx.y * 4 + wave;
  const int lm = lane & 15, kh = (lane >> 4) * 2, khalf = lane >> 4;
  const int m0 = tm * 32 + lm, m1 = m0 + 16;
  const int n0 = tn * 32 + lm, n1 = n0 + 16;

  const float* __restrict__ A =
      zin + (use_b ? (size_t)b * G_ * N_ : (size_t)bp * zin_stride);
  const float* __restrict__ Bm = aij + (size_t)bp * N_ * N_;

  v8f c00 = {}, c01 = {}, c10 = {}, c11 = {};
  for (int kk = 0; kk < N_; kk += 4) {
    const int k0 = kk + kh;
    v2f a0; a0.x = A[(size_t)m0 * N_ + k0]; a0.y = A[(size_t)m0 * N_ + k0 + 1];
    v2f a1; a1.x = A[(size_t)m1 * N_ + k0]; a1.y = A[(size_t)m1 * N_ + k0 + 1];
    v2f b0; b0.x = Bm[(size_t)k0 * N_ + n0]; b0.y = Bm[(size_t)(k0 + 1) * N_ + n0];
    v2f b1; b1.x = Bm[(size_t)k0 * N_ + n1]; b1.y = Bm[(size_t)(k0 + 1) * N_ + n1];
    c00 = wmma4(a0, b0, c00); c01 = wmma4(a0, b1, c01);
    c10 = wmma4(a1, b0, c10); c11 = wmma4(a1, b1, c11);
  }
  float* __restrict__ C = zout + (size_t)bp * zout_stride;
#pragma unroll
  for (int v = 0; v < 8; ++v) {
    const int r0 = tm * 32 + v + 8 * khalf, r1 = r0 + 16;
    C[(size_t)r0 * N_ + n0] = c00[v];
    C[(size_t)r0 * N_ + n1] = c01[v];
    C[(size_t)r1 * N_ + n0] = c10[v];
    C[(size_t)r1 * N_ + n1] = c11[v];
  }
}

// ---------------------------------------------------------------------------
// Kernel 5: out[b, p*F+f, n] = relu( h_p(F x 384) @ Z(384 x N) + bias[f] )
// Z row kg: k = kg>>7 selects x (k==0) or workspace tap (k-1); g = kg&127
// 32x32 block per wave. grid(4, 4, 64) block 128
// ---------------------------------------------------------------------------
__global__ __launch_bounds__(128) void k_final(const float* __restrict__ x,
                                               const float* __restrict__ zws,
                                               const float* __restrict__ h,
                                               const float* __restrict__ bias,
                                               float* __restrict__ out) {
  const int lane = threadIdx.x & 31, wave = threadIdx.x >> 5;
  const int bp = blockIdx.z, b = bp >> 2, p = bp & 3;
  const int tm = blockIdx.x, tn = blockIdx.y * 4 + wave;
  const int lm = lane & 15, kh = (lane >> 4) * 2, khalf = lane >> 4;
  const int m0 = tm * 32 + lm, m1 = m0 + 16;
  const int n0 = tn * 32 + lm, n1 = n0 + 16;
  const int KG = KH_ * G_; // 384

  const float* __restrict__ A = h + (size_t)p * F_ * KG;  // (F, 384) row-major

  v8f c00 = {}, c01 = {}, c10 = {}, c11 = {};
  for (int kk = 0; kk < KG; kk += 4) {
    const int k0 = kk + kh;
    v2f a0; a0.x = A[(size_t)m0 * KG + k0]; a0.y = A[(size_t)m0 * KG + k0 + 1];
    v2f a1; a1.x = A[(size_t)m1 * KG + k0]; a1.y = A[(size_t)m1 * KG + k0 + 1];

    const int ki0 = k0 >> 7, g0 = k0 & 127;
    const int ki1 = (k0 + 1) >> 7, g1 = (k0 + 1) & 127;
    const float* r0 = (ki0 == 0) ? x + ((size_t)b * G_ + g0) * N_
                                 : zws + (((size_t)bp * 2 + (ki0 - 1)) * G_ + g0) * N_;
    const float* r1 = (ki1 == 0) ? x + ((size_t)b * G_ + g1) * N_
                                 : zws + (((size_t)bp * 2 + (ki1 - 1)) * G_ + g1) * N_;
    v2f b0; b0.x = r0[n0]; b0.y = r1[n0];
    v2f b1; b1.x = r0[n1]; b1.y = r1[n1];
    c00 = wmma4(a0, b0, c00); c01 = wmma4(a0, b1, c01);
    c10 = wmma4(a1, b0, c10); c11 = wmma4(a1, b1, c11);
  }
#pragma unroll
  for (int v = 0; v < 8; ++v) {
    const int f0 = tm * 32 + v + 8 * khalf, f1 = f0 + 16;
    const float bv0 = bias[f0], bv1 = bias[f1];
    out[(((size_t)b * P_ + p) * F_ + f0) * N_ + n0] = fmaxf(c00[v] + bv0, 0.f);
    out[(((size_t)b * P_ + p) * F_ + f0) * N_ + n1] = fmaxf(c01[v] + bv0, 0.f);
    out[(((size_t)b * P_ + p) * F_ + f1) * N_ + n0] = fmaxf(c10[v] + bv1, 0.f);
    out[(((size_t)b * P_ + p) * F_ + f1) * N_ + n1] = fmaxf(c11[v] + bv1, 0.f);
  }
}

// ---------------------------------------------------------------------------
// Workspace layout (floats):  wx[BP*G*N] | aij[BP*N*N] | z[BP*2*G*N]
// = 4,194,304 + 16,777,216 + 8,388,608 floats = 112 MiB  (ws_size must cover)
// ---------------------------------------------------------------------------
extern "C" void kernel_launch(void* const* d_in, const int* in_sizes, int n_in,
                              void* d_out, int out_size, void* d_ws, size_t ws_size,
                              hipStream_t stream) {
  const float* x     = (const float*)d_in[0];
  const float* S     = (const float*)d_in[1];
  const float* W     = (const float*)d_in[2];
  // d_in[3] = mixer (unused in key-query mode)
  const float* h     = (const float*)d_in[4];
  const float* bias  = (const float*)d_in[5];
  float* out = (float*)d_out;

  float* ws  = (float*)d_ws;
  float* wx  = ws;
  float* aij = wx  + (size_t)BP_ * G_ * N_;
  float* z   = aij + (size_t)BP_ * N_ * N_;
  const long GN = (long)G_ * N_;

  dim3 blk(128);
  // 1) Wx = W[p] @ x[b]
  k_wx<<<dim3(G_ / 32, N_ / 32 / 4, BP_), blk, 0, stream>>>(x, W, wx);
  // 2) scores = mask(leakyrelu(x^T @ Wx))
  k_scores<<<dim3(N_ / 32, N_ / 32 / 4, BP_), blk, 0, stream>>>(x, wx, S, aij);
  // 3) row softmax (+ mask zeroing), in place
  k_softmax<<<dim3(BP_ * N_), dim3(256), 0, stream>>>(aij, S);
  // 4) z1 = x @ aij     (tap slot 0)
  k_hop<<<dim3(G_ / 32, N_ / 32 / 4, BP_), blk, 0, stream>>>(x, 0, 1, aij, z, 2 * GN);
  // 5) z2 = z1 @ aij    (tap slot 1)
  k_hop<<<dim3(G_ / 32, N_ / 32 / 4, BP_), blk, 0, stream>>>(z, 2 * GN, 0, aij, z + GN, 2 * GN);
  // 6) y = relu(h @ [x;z1;z2] + b) -> (B, P*F, N)
  k_final<<<dim3(F_ / 32, N_ / 32 / 4, BP_), blk, 0, stream>>>(x, z, h, bias, out);
}